// CompressedSensingInception_69733089017894
// MI455X (gfx1250) — compile-verified
//
#include <hip/hip_runtime.h>

typedef float v2f __attribute__((ext_vector_type(2)));
typedef float v8f __attribute__((ext_vector_type(8)));

constexpr int NB   = 64;     // batch
constexpr int S    = 5184;   // 72*72 sparse rows
constexpr int NCOL = 192;    // NB*3 columns
constexpr int MP   = 96;     // 81 padded to 96 (multiple of 16)

__device__ __forceinline__ float lrelu(float x) { return x > 0.f ? x : 0.2f * x; }

// ---------------------------------------------------------------------------
// Prep: fragment-packed layouts so every WMMA operand is one b64 load.
//  matPF: step-1 A.  fragment f=k/4, kh=(k/2)&1, slot=k&1:
//         matPF[((f*2+kh)*MP + m)*2 + slot] = mat[k][m]
//  matTF: step-2 A.  k-dim is the 96 axis: f2=m/4, kh2=(m/2)&1, slot2=m&1:
//         matTF[((f2*2+kh2)*S + s)*2 + slot2] = mat[s][m]
// ---------------------------------------------------------------------------
__global__ void k_prep(const float* __restrict__ mat, float* __restrict__ matPF,
                       float* __restrict__ matTF) {
  int idx = blockIdx.x * 256 + threadIdx.x;
  if (idx >= S * MP) return;
  int s = idx / MP, m = idx % MP;
  float v = (m < 81) ? mat[s * 81 + m] : 0.f;
  int f = s >> 2, kh = (s >> 1) & 1, sl = s & 1;
  matPF[(((f * 2 + kh) * MP) + m) * 2 + sl] = v;
  int f2 = m >> 2, kh2 = (m >> 1) & 1, sl2 = m & 1;
  matTF[(((f2 * 2 + kh2) * S) + s) * 2 + sl2] = v;
}

// ---------------------------------------------------------------------------
// FISTA: 12 blocks x 16 columns, 100 iterations, f32 WMMA 16x16x4.
// bufA (k-pair interleaved [k/2][n][2]) carries y_mom; bufB (flat [s][n])
// carries y_new (y_last / final output consumed by the conv path).
// ---------------------------------------------------------------------------
__global__ __launch_bounds__(256) void k_fista(
    const float* __restrict__ x, const float* __restrict__ matPF,
    const float* __restrict__ matTF, float* __restrict__ bufA,
    float* __restrict__ bufB) {
  const int n0    = blockIdx.x * 16;
  const int tid   = threadIdx.x;
  const int wave  = tid >> 5;
  const int lane  = tid & 31;
  const int l16   = lane & 15;
  const int khalf = lane >> 4;   // 0: K 0..1, 1: K 2..3 within a 16x16x4 step

  __shared__ float Qs[MP * 16];   // Q = Im - P, pair-interleaved [m/2][j][2]
  __shared__ float Ims[MP * 16];  // flat [m][j]

  // stage Im for our 16 columns (rows >= 81 are zero padding)
  for (int idx = tid; idx < MP * 16; idx += 256) {
    int p = idx >> 4, j = idx & 15, n = n0 + j;
    Ims[idx] = (p < 81) ? x[(n / 3) * 243 + p * 3 + (n % 3)] : 0.f;
  }
  // y0 = 0 for both carries
  for (int idx = tid; idx < S * 16; idx += 256) {
    int srow = idx >> 4, col = n0 + (idx & 15);
    bufA[((srow >> 1) * NCOL + col) * 2 + (srow & 1)] = 0.f;
    bufB[srow * NCOL + col] = 0.f;
  }
  __syncthreads();

  const v2f* Ap  = (const v2f*)matPF;
  const v2f* At  = (const v2f*)matTF;
  const v2f* Yp  = (const v2f*)bufA;
  const v2f* Qp  = (const v2f*)Qs;

  float t = 1.0f;
  for (int it = 0; it < 100; ++it) {
    // ---- step 1: P = M^T * Y  (six 16x16 tiles over waves 0..5) ----
    if (wave < 6) {
      v8f acc = {0.f, 0.f, 0.f, 0.f, 0.f, 0.f, 0.f, 0.f};
      const int mg = wave * 16 + l16;   // output row (0..95)
      // fragment pointers; per k-step stride: A 2*MP v2f, B 2*NCOL v2f
      const v2f* ap = Ap + khalf * MP + mg;
      const v2f* bp = Yp + khalf * NCOL + n0 + l16;
      v2f a0 = *ap, b0 = *bp;
      for (int f = 0; f < S / 4 - 1; ++f) {   // software-pipelined
        ap += 2 * MP;
        bp += 2 * NCOL;
        v2f a1 = *ap;
        v2f b1 = *bp;
        acc = __builtin_amdgcn_wmma_f32_16x16x4_f32(false, a0, false, b0,
                                                    (short)0, acc, false, false);
        a0 = a1; b0 = b1;
      }
      acc = __builtin_amdgcn_wmma_f32_16x16x4_f32(false, a0, false, b0,
                                                  (short)0, acc, false, false);
#pragma unroll
      for (int r = 0; r < 8; ++r) {
        int m = wave * 16 + r + 8 * khalf;
        // Q = Im - P, pair-interleaved for b64 LDS reads in step 2
        Qs[((m >> 1) * 16 + l16) * 2 + (m & 1)] = Ims[m * 16 + l16] - acc[r];
      }
    }
    __syncthreads();

    float tn = 0.5f * (1.0f + sqrtf(1.0f + 4.0f * t * t));
    float mo = (t - 1.0f) / tn;

    // ---- step 2: R = M * Q, fused soft-threshold + momentum ----
    for (int rt = wave; rt < S / 16; rt += 8) {
      const int row0 = rt * 16;
      v8f acc = {0.f, 0.f, 0.f, 0.f, 0.f, 0.f, 0.f, 0.f};
      const v2f* ap = At + khalf * S + row0 + l16;
      const v2f* qp = Qp + khalf * 16 + l16;
#pragma unroll
      for (int f = 0; f < MP / 4; ++f) {
        v2f a  = ap[f * 2 * S];
        v2f bb = qp[f * 2 * 16];
        acc = __builtin_amdgcn_wmma_f32_16x16x4_f32(false, a, false, bb,
                                                    (short)0, acc, false, false);
      }
#pragma unroll
      for (int r = 0; r < 8; ++r) {
        int srow = row0 + r + 8 * khalf;
        int col  = n0 + l16;
        int offA = ((srow >> 1) * NCOL + col) * 2 + (srow & 1);
        int offB = srow * NCOL + col;
        float ytmp  = bufA[offA];
        float ylast = bufB[offB];
        float w     = ytmp + acc[r];                       // MU = 1
        float ynew  = fmaxf(w - 0.005f, 0.f) - fmaxf(-w - 0.005f, 0.f);
        bufA[offA] = ynew + mo * (ynew - ylast);           // y_mom
        bufB[offB] = ynew;                                 // y_last / final
      }
    }
    t = tn;
    __syncthreads();
  }
}

// ---------------------------------------------------------------------------
// bn_x stats over the reflect-padded xi (B,73,73,3): deterministic 1-block.
// ---------------------------------------------------------------------------
__global__ void k_bnx_stats(const float* __restrict__ bufB, float* __restrict__ stats) {
  __shared__ float rs[256], rq[256];
  int tid = threadIdx.x;
  const int NPIX = NB * 73 * 73;
  for (int c = 0; c < 3; ++c) {
    float s = 0.f, q = 0.f;
    for (int p = tid; p < NPIX; p += 256) {
      int b = p / (73 * 73), r = p % (73 * 73);
      int i = r / 73, j = r % 73;
      int si = (i == 0) ? 1 : i - 1;
      int sj = (j == 0) ? 1 : j - 1;
      float v = bufB[(si * 72 + sj) * NCOL + b * 3 + c];
      s += v; q += v * v;
    }
    rs[tid] = s; rq[tid] = q; __syncthreads();
    for (int st = 128; st > 0; st >>= 1) {
      if (tid < st) { rs[tid] += rs[tid + st]; rq[tid] += rq[tid + st]; }
      __syncthreads();
    }
    if (tid == 0) { stats[c * 2] = rs[0]; stats[c * 2 + 1] = rq[0]; }
    __syncthreads();
  }
}

// Generic deterministic per-channel sum/sumsq for [rows][nch] tensors.
__global__ void k_stats(const float* __restrict__ src, int rows, int nch,
                        float* __restrict__ outp) {
  __shared__ float rs[256], rq[256];
  int tid = threadIdx.x;
  for (int c = 0; c < nch; ++c) {
    float s = 0.f, q = 0.f;
    for (int r = tid; r < rows; r += 256) {
      float v = src[r * nch + c];
      s += v; q += v * v;
    }
    rs[tid] = s; rq[tid] = q; __syncthreads();
    for (int st = 128; st > 0; st >>= 1) {
      if (tid < st) { rs[tid] += rs[tid + st]; rq[tid] += rq[tid + st]; }
      __syncthreads();
    }
    if (tid == 0) { outp[c * 2] = rs[0]; outp[c * 2 + 1] = rq[0]; }
    __syncthreads();
  }
}

// ---------------------------------------------------------------------------
// conv 5x5 VALID on BN(reflect-padded xi): (B,73,73,3) -> (B,69,69,8)
// ---------------------------------------------------------------------------
__global__ void k_conv5(const float* __restrict__ bufB, const float* __restrict__ stats,
                        const float* __restrict__ bn_g, const float* __restrict__ bn_b,
                        const float* __restrict__ w5, const float* __restrict__ b5,
                        float* __restrict__ outp) {
  int idx = blockIdx.x * 256 + threadIdx.x;
  const int TOT = NB * 69 * 69 * 8;
  if (idx >= TOT) return;
  int o = idx & 7, t = idx >> 3;
  int oj = t % 69; t /= 69;
  int oi = t % 69; int b = t / 69;
  const float cnt = (float)(NB * 73 * 73);
  float sc[3], sh[3];
  for (int c = 0; c < 3; ++c) {
    float m = stats[c * 2] / cnt;
    float v = stats[c * 2 + 1] / cnt - m * m;
    sc[c] = bn_g[c] * rsqrtf(v + 1e-3f);
    sh[c] = bn_b[c] - sc[c] * m;
  }
  float acc = b5[o];
  for (int ki = 0; ki < 5; ++ki)
    for (int kj = 0; kj < 5; ++kj) {
      int i = oi + ki, j = oj + kj;
      int si = (i == 0) ? 1 : i - 1;
      int sj = (j == 0) ? 1 : j - 1;
      const float* px = &bufB[(si * 72 + sj) * NCOL + b * 3];
      for (int c = 0; c < 3; ++c)
        acc += (sc[c] * px[c] + sh[c]) * w5[((ki * 5 + kj) * 3 + c) * 8 + o];
    }
  outp[idx] = acc;
}

// maxpool 4x4 stride 4 SAME: (B,69,69,8) -> (B,18,18,8); pad lo=1
__global__ void k_pool4(const float* __restrict__ c5, float* __restrict__ outp) {
  int idx = blockIdx.x * 256 + threadIdx.x;
  const int TOT = NB * 18 * 18 * 8;
  if (idx >= TOT) return;
  int o = idx & 7, t = idx >> 3;
  int s = t % 18; t /= 18;
  int r = t % 18; int b = t / 18;
  float m = -__builtin_inff();
  for (int i = 4 * r - 1; i <= 4 * r + 2; ++i) {
    if (i < 0 || i >= 69) continue;
    for (int j = 4 * s - 1; j <= 4 * s + 2; ++j) {
      if (j < 0 || j >= 69) continue;
      m = fmaxf(m, c5[((b * 69 + i) * 69 + j) * 8 + o]);
    }
  }
  outp[idx] = m;
}

// 1x1 conv (8->2) + relu, then maxpool 2x2 -> write out channels 1..2
__global__ void k_xfinal(const float* __restrict__ p4, const float* __restrict__ wx2,
                         const float* __restrict__ bx2, float* __restrict__ outp) {
  int idx = blockIdx.x * 256 + threadIdx.x;
  const int TOT = NB * 9 * 9 * 2;
  if (idx >= TOT) return;
  int o2 = idx & 1, t = idx >> 1;
  int q = t % 9; t /= 9;
  int p = t % 9; int b = t / 9;
  float m = -__builtin_inff();
  for (int r = 2 * p; r <= 2 * p + 1; ++r)
    for (int s = 2 * q; s <= 2 * q + 1; ++s) {
      float a = bx2[o2];
      for (int o = 0; o < 8; ++o)
        a += p4[((b * 18 + r) * 18 + s) * 8 + o] * wx2[o * 2 + o2];
      m = fmaxf(m, fmaxf(a, 0.f));
    }
  outp[((b * 9 + p) * 9 + q) * 5 + 1 + o2] = m;
}

// y path: 1x1 conv + relu
__global__ void k_y1(const float* __restrict__ x, const float* __restrict__ wy1,
                     const float* __restrict__ by1, float* __restrict__ y1) {
  int idx = blockIdx.x * 256 + threadIdx.x;
  if (idx >= NB * 81) return;
  const float* px = &x[idx * 3];
  y1[idx] = fmaxf(by1[0] + px[0] * wy1[0] + px[1] * wy1[1] + px[2] * wy1[2], 0.f);
}

// y path: 7x7 SAME conv
__global__ void k_y2(const float* __restrict__ y1, const float* __restrict__ wy7,
                     const float* __restrict__ by7, float* __restrict__ y2) {
  int idx = blockIdx.x * 256 + threadIdx.x;
  if (idx >= NB * 81) return;
  int p = idx % 81, b = idx / 81;
  int i = p / 9, j = p % 9;
  float a = by7[0];
  for (int ki = 0; ki < 7; ++ki) {
    int ii = i + ki - 3;
    if (ii < 0 || ii >= 9) continue;
    for (int kj = 0; kj < 7; ++kj) {
      int jj = j + kj - 3;
      if (jj < 0 || jj >= 9) continue;
      a += y1[b * 81 + ii * 9 + jj] * wy7[ki * 7 + kj];
    }
  }
  y2[idx] = a;
}

// z path: down conv 1 (pre-BN): (B,9,9,3) -> (B,3,3,12)
__global__ void k_z1(const float* __restrict__ x, const float* __restrict__ wd1,
                     float* __restrict__ z1p) {
  int idx = blockIdx.x * 256 + threadIdx.x;
  const int TOT = NB * 3 * 3 * 12;
  if (idx >= TOT) return;
  int o = idx % 12, t = idx / 12;
  int q = t % 3; t /= 3;
  int p = t % 3; int b = t / 3;
  float a = 0.f;
  for (int ki = 0; ki < 3; ++ki)
    for (int kj = 0; kj < 3; ++kj) {
      const float* px = &x[((b * 9 + 3 * p + ki) * 9 + 3 * q + kj) * 3];
      for (int c = 0; c < 3; ++c) a += px[c] * wd1[((ki * 3 + kj) * 3 + c) * 12 + o];
    }
  z1p[idx] = a;
}

// z path: down conv 2 on leaky(bn(z1)): -> (B,1,1,24)
__global__ void k_z2(const float* __restrict__ z1p, const float* __restrict__ stats,
                     const float* __restrict__ g1, const float* __restrict__ b1,
                     const float* __restrict__ wd2, float* __restrict__ z2p) {
  int idx = blockIdx.x * 256 + threadIdx.x;
  if (idx >= NB * 24) return;
  int o = idx % 24, b = idx / 24;
  float a = 0.f;
  for (int ki = 0; ki < 3; ++ki)
    for (int kj = 0; kj < 3; ++kj)
      for (int c = 0; c < 12; ++c) {
        float v   = z1p[((b * 3 + ki) * 3 + kj) * 12 + c];
        float m   = stats[8 + c * 2] / 576.f;
        float var = stats[8 + c * 2 + 1] / 576.f - m * m;
        v = lrelu(g1[c] * (v - m) * rsqrtf(var + 1e-3f) + b1[c]);
        a += v * wd2[((ki * 3 + kj) * 12 + c) * 24 + o];
      }
  z2p[idx] = a;
}

// z path: conv_transpose 1 on leaky(bn(z2)): (B,1,1,24) -> (B,3,3,12)
// out[i][j][o] = sum_c act(z2[c]) * wu1[2-i][2-j][c][o]
__global__ void k_z3(const float* __restrict__ z2p, const float* __restrict__ stats,
                     const float* __restrict__ g2, const float* __restrict__ b2,
                     const float* __restrict__ wu1, float* __restrict__ z3p) {
  int idx = blockIdx.x * 256 + threadIdx.x;
  const int TOT = NB * 3 * 3 * 12;
  if (idx >= TOT) return;
  int o = idx % 12, t = idx / 12;
  int j = t % 3; t /= 3;
  int i = t % 3; int b = t / 3;
  int ki = 2 - i, kj = 2 - j;
  float a = 0.f;
  for (int c = 0; c < 24; ++c) {
    float v   = z2p[b * 24 + c];
    float m   = stats[32 + c * 2] / 64.f;
    float var = stats[32 + c * 2 + 1] / 64.f - m * m;
    v = lrelu(g2[c] * (v - m) * rsqrtf(var + 1e-3f) + b2[c]);
    a += v * wu1[((ki * 3 + kj) * 24 + c) * 12 + o];
  }
  z3p[idx] = a;
}

// final: w path (ch0), y BN (ch3), z conv_transpose 2 + relu (ch4)
__global__ void k_final(const float* __restrict__ x, const float* __restrict__ stats,
                        const float* __restrict__ ww1, const float* __restrict__ bw1,
                        const float* __restrict__ y2v, const float* __restrict__ gy,
                        const float* __restrict__ byb, const float* __restrict__ z3p,
                        const float* __restrict__ gu1, const float* __restrict__ bu1,
                        const float* __restrict__ z1p, const float* __restrict__ g1,
                        const float* __restrict__ b1, const float* __restrict__ wu2,
                        float* __restrict__ outp) {
  int idx = blockIdx.x * 256 + threadIdx.x;
  if (idx >= NB * 81) return;
  int b = idx / 81, p = idx % 81;
  int i = p / 9, j = p % 9;
  float* po = &outp[idx * 5];
  // ch0: w path
  const float* px = &x[idx * 3];
  po[0] = fmaxf(bw1[0] + px[0] * ww1[0] + px[1] * ww1[1] + px[2] * ww1[2], 0.f);
  // ch3: y path BN
  {
    float m   = stats[6] / 5184.f;
    float var = stats[7] / 5184.f - m * m;
    po[3] = gy[0] * (y2v[idx] - m) * rsqrtf(var + 1e-3f) + byb[0];
  }
  // ch4: z path final transpose conv + relu over concat([relu(bn(z3)), leaky(bn(z1))])
  {
    int u = i / 3, v = j / 3;
    int ki = 2 - (i % 3), kj = 2 - (j % 3);
    int base = ((b * 3 + u) * 3 + v) * 12;
    float a = 0.f;
    for (int c = 0; c < 12; ++c) {
      float z   = z3p[base + c];
      float m   = stats[80 + c * 2] / 576.f;
      float var = stats[80 + c * 2 + 1] / 576.f - m * m;
      z = fmaxf(gu1[c] * (z - m) * rsqrtf(var + 1e-3f) + bu1[c], 0.f);
      a += z * wu2[(ki * 3 + kj) * 24 + c];
    }
    for (int c = 0; c < 12; ++c) {
      float z   = z1p[base + c];
      float m   = stats[8 + c * 2] / 576.f;
      float var = stats[8 + c * 2 + 1] / 576.f - m * m;
      z = lrelu(g1[c] * (z - m) * rsqrtf(var + 1e-3f) + b1[c]);
      a += z * wu2[(ki * 3 + kj) * 24 + 12 + c];
    }
    po[4] = fmaxf(a, 0.f);
  }
}

// ---------------------------------------------------------------------------
extern "C" void kernel_launch(void* const* d_in, const int* in_sizes, int n_in,
                              void* d_out, int out_size, void* d_ws, size_t ws_size,
                              hipStream_t stream) {
  (void)in_sizes; (void)n_in; (void)out_size; (void)ws_size;
  const float* x      = (const float*)d_in[0];
  const float* mat    = (const float*)d_in[1];
  const float* bn_x_g = (const float*)d_in[2];
  const float* bn_x_b = (const float*)d_in[3];
  const float* w5     = (const float*)d_in[4];
  const float* b5     = (const float*)d_in[5];
  const float* wx2    = (const float*)d_in[6];
  const float* bx2    = (const float*)d_in[7];
  const float* wy1    = (const float*)d_in[8];
  const float* by1    = (const float*)d_in[9];
  const float* wy7    = (const float*)d_in[10];
  const float* by7    = (const float*)d_in[11];
  const float* bn_y_g = (const float*)d_in[12];
  const float* bn_y_b = (const float*)d_in[13];
  const float* ww1    = (const float*)d_in[14];
  const float* bw1    = (const float*)d_in[15];
  const float* wd1    = (const float*)d_in[16];
  const float* bnd1_g = (const float*)d_in[17];
  const float* bnd1_b = (const float*)d_in[18];
  const float* wd2    = (const float*)d_in[19];
  const float* bnd2_g = (const float*)d_in[20];
  const float* bnd2_b = (const float*)d_in[21];
  const float* wu1    = (const float*)d_in[22];
  const float* bnu1_g = (const float*)d_in[23];
  const float* bnu1_b = (const float*)d_in[24];
  const float* wu2    = (const float*)d_in[25];
  float* outp = (float*)d_out;

  float* ws     = (float*)d_ws;
  float* matPF  = ws;                        // 5184*96
  float* matTF  = matPF + S * MP;            // 5184*96
  float* bufA   = matTF + S * MP;            // 5184*192 (pair-interleaved)
  float* bufB   = bufA + S * NCOL;           // 5184*192 (flat [s][n])
  float* conv5o = bufB + S * NCOL;           // 64*69*69*8
  float* pool4o = conv5o + NB * 69 * 69 * 8; // 64*18*18*8
  float* y1v    = pool4o + NB * 18 * 18 * 8; // 5184
  float* y2v    = y1v + NB * 81;             // 5184
  float* z1p    = y2v + NB * 81;             // 6912
  float* z2p    = z1p + NB * 9 * 12;         // 1536
  float* z3p    = z2p + NB * 24;             // 6912
  float* stats  = z3p + NB * 9 * 12;         // 104 floats of BN sums

  k_prep<<<(S * MP + 255) / 256, 256, 0, stream>>>(mat, matPF, matTF);
  k_fista<<<NCOL / 16, 256, 0, stream>>>(x, matPF, matTF, bufA, bufB);

  k_bnx_stats<<<1, 256, 0, stream>>>(bufB, stats);
  k_conv5<<<(NB * 69 * 69 * 8 + 255) / 256, 256, 0, stream>>>(bufB, stats, bn_x_g,
                                                              bn_x_b, w5, b5, conv5o);
  k_pool4<<<(NB * 18 * 18 * 8 + 255) / 256, 256, 0, stream>>>(conv5o, pool4o);
  k_xfinal<<<(NB * 81 * 2 + 255) / 256, 256, 0, stream>>>(pool4o, wx2, bx2, outp);

  k_y1<<<(NB * 81 + 255) / 256, 256, 0, stream>>>(x, wy1, by1, y1v);
  k_y2<<<(NB * 81 + 255) / 256, 256, 0, stream>>>(y1v, wy7, by7, y2v);
  k_stats<<<1, 256, 0, stream>>>(y2v, NB * 81, 1, stats + 6);

  k_z1<<<(NB * 9 * 12 + 255) / 256, 256, 0, stream>>>(x, wd1, z1p);
  k_stats<<<1, 256, 0, stream>>>(z1p, NB * 9, 12, stats + 8);
  k_z2<<<(NB * 24 + 255) / 256, 256, 0, stream>>>(z1p, stats, bnd1_g, bnd1_b, wd2, z2p);
  k_stats<<<1, 256, 0, stream>>>(z2p, NB, 24, stats + 32);
  k_z3<<<(NB * 9 * 12 + 255) / 256, 256, 0, stream>>>(z2p, stats, bnd2_g, bnd2_b, wu1, z3p);
  k_stats<<<1, 256, 0, stream>>>(z3p, NB * 9, 12, stats + 80);

  k_final<<<(NB * 81 + 255) / 256, 256, 0, stream>>>(
      x, stats, ww1, bw1, y2v, bn_y_g, bn_y_b, z3p, bnu1_g, bnu1_b, z1p, bnd1_g,
      bnd1_b, wu2, outp);
}